// LearnableGraphSparsifier_79783312491206
// MI455X (gfx1250) — compile-verified
//
#include <hip/hip_runtime.h>
#include <hip/hip_bf16.h>

// ---- problem constants (from reference) ------------------------------------
constexpr int N_NODES   = 1000;
constexpr int N_EDGES   = 8000;
constexpr int BATCH     = 64;
constexpr int IN_DIM    = 64;
constexpr int HID       = 128;
constexpr int N_CLASSES = 2;
constexpr int NT        = BATCH * N_NODES;        // 64000 total nodes
constexpr int ET        = BATCH * N_EDGES;        // 512000 total edges
constexpr int HN        = NT * HID;               // 8,192,000 h elements
constexpr int KREAD     = N_NODES * HID;          // 128000 readout K

typedef __attribute__((ext_vector_type(2))) float v2f;
typedef __attribute__((ext_vector_type(8))) float v8f;
typedef int vi4 __attribute__((__vector_size__(16)));   // matches builtin proto

// ---- workspace layout (floats) ---------------------------------------------
constexpr size_t WS_DEG  = 0;
constexpr size_t WS_DINV = (size_t)NT;
constexpr size_t WS_H1   = (size_t)2 * NT;
constexpr size_t WS_HACC = WS_H1 + (size_t)HN;
constexpr size_t WS_GACC = WS_HACC + (size_t)HN;

// ---- CDNA5 async global->LDS copy (ASYNCcnt path), with safe fallback ------
#if __has_builtin(__builtin_amdgcn_global_load_async_to_lds_b128)
#define HAVE_ASYNC_LDS 1
#endif

__device__ __forceinline__ void async_copy_b128(const float* g, float* l) {
#ifdef HAVE_ASYNC_LDS
    __builtin_amdgcn_global_load_async_to_lds_b128(
        (__attribute__((address_space(1))) vi4*)g,
        (__attribute__((address_space(3))) vi4*)l,
        0, 0);
#else
    *reinterpret_cast<float4*>(l) = *reinterpret_cast<const float4*>(g);
#endif
}

__device__ __forceinline__ void async_join() {
#ifdef HAVE_ASYNC_LDS
#if __has_builtin(__builtin_amdgcn_s_wait_asynccnt)
    __builtin_amdgcn_s_wait_asynccnt(0);
#else
    asm volatile("s_wait_asynccnt 0x0" ::: "memory");
#endif
#endif
}

// ---- 1) ew sparsify / ni clamp / init deg & gacc ---------------------------
__global__ void prep_kernel(const float* __restrict__ ew_in,
                            const float* __restrict__ ni_in,
                            float* __restrict__ out_ew,
                            float* __restrict__ out_ni,
                            float* __restrict__ deg,
                            float* __restrict__ gacc) {
    int i = blockIdx.x * blockDim.x + threadIdx.x;
    if (i < N_EDGES) {
        float w = ew_in[i];
        out_ew[i] = (w < 0.2f) ? 0.0f : fminf(w, 1.0f);
    }
    if (i < N_NODES) {
        out_ni[i] = fminf(fmaxf(ni_in[i], 0.0f), 1.0f);
    }
    if (i < NT) deg[i] = 1.0f;                 // self-loop weight
    if (i < BATCH * 32) gacc[i] = 0.0f;        // readout accumulator
}

// ---- 2) degree accumulation over edges -------------------------------------
__global__ void deg_kernel(const int* __restrict__ src,
                           const int* __restrict__ dst,
                           const float* __restrict__ ew,
                           float* __restrict__ deg) {
    int e = blockIdx.x * blockDim.x + threadIdx.x;
    if (e < ET) {
        float w = ew[e % N_EDGES];
        atomicAdd(&deg[dst[e]], w);
    }
}

// ---- 3) symmetric norm factors ---------------------------------------------
__global__ void dinv_kernel(const float* __restrict__ deg,
                            float* __restrict__ dinv) {
    int i = blockIdx.x * blockDim.x + threadIdx.x;
    if (i < NT) {
        dinv[i] = rsqrtf(fmaxf(deg[i], 1e-12f));   // deg >= 1 (self-loop)
    }
}

// ---- 4) GEMM1: h1[NT,HID] = (x @ W1) * ni_raw (scale commuted to epilogue) -
// Block = 256 threads (8 waves), M-tile = 64 rows, all 128 N columns.
// x-tile (16 KB) and W1 (32 KB) staged in LDS via async global->LDS copies.
// Inner loop is pure ds_load + v_wmma_f32_16x16x4_f32; per-row ni scaling
// and the self-loop seed (h1 * dinv^2 into hacc) happen in the epilogue.
constexpr int G1_MT = 64;
__global__ void __launch_bounds__(256)
gemm1_wmma(const float* __restrict__ x,     // [NT, IN_DIM]
           const float* __restrict__ ni,    // [N_NODES] raw
           const float* __restrict__ W1,    // [IN_DIM, HID]
           const float* __restrict__ dinv,  // [NT]
           float* __restrict__ h1,          // [NT, HID]
           float* __restrict__ hacc) {      // [NT, HID]
    __shared__ float xs[G1_MT * IN_DIM];    // 16 KB
    __shared__ float wsh[IN_DIM * HID];     // 32 KB

    const int tid = threadIdx.y * 32 + threadIdx.x;
    const int m0  = blockIdx.x * G1_MT;

    #pragma unroll
    for (int i = tid; i < (G1_MT * IN_DIM) / 4; i += 256)
        async_copy_b128(x + (size_t)m0 * IN_DIM + i * 4, xs + i * 4);
    #pragma unroll
    for (int i = tid; i < (IN_DIM * HID) / 4; i += 256)
        async_copy_b128(W1 + i * 4, wsh + i * 4);
    async_join();
    __syncthreads();

    const int lane = threadIdx.x;
    const int wave = threadIdx.y;            // 0..7 -> N tile
    const int n0   = wave * 16;
    const int l16  = lane & 15;
    const int half = lane >> 4;              // K pair select

    v8f c[4] = {};
    #pragma unroll
    for (int k0 = 0; k0 < IN_DIM; k0 += 4) {
        const int kb = k0 + half * 2;
        v2f b;
        b.x = wsh[kb * HID + n0 + l16];          // B: K=kb,   N=l16
        b.y = wsh[(kb + 1) * HID + n0 + l16];    //    K=kb+1
        #pragma unroll
        for (int mt = 0; mt < 4; ++mt) {
            const float2 av =
                *reinterpret_cast<const float2*>(&xs[(mt * 16 + l16) * IN_DIM + kb]);
            v2f a;
            a.x = av.x;
            a.y = av.y;
            c[mt] = __builtin_amdgcn_wmma_f32_16x16x4_f32(
                        false, a, false, b, (short)0, c[mt], false, false);
        }
    }

    #pragma unroll
    for (int mt = 0; mt < 4; ++mt) {
        #pragma unroll
        for (int v = 0; v < 8; ++v) {
            const int M = m0 + mt * 16 + v + half * 8;
            const int N = n0 + l16;
            const float s   = ni[M % N_NODES];   // diag(ni)·(X·W) == (ni·X)·W
            const float dv  = dinv[M];
            const float val = c[mt][v] * s;
            h1[(size_t)M * HID + N]   = val;
            hacc[(size_t)M * HID + N] = val * dv * dv;   // self-loop term
        }
    }
}

// ---- 5) edge scatter: hacc[dst] += h1[src] * norm --------------------------
__global__ void scatter_kernel(const int* __restrict__ src,
                               const int* __restrict__ dst,
                               const float* __restrict__ ew,
                               const float* __restrict__ dinv,
                               const float* __restrict__ h1,
                               float* __restrict__ hacc) {
    const int t  = blockIdx.x * blockDim.x + threadIdx.x;
    const int e  = t >> 5;
    const int j4 = (t & 31) * 4;
    if (e >= ET) return;
    const int s = src[e], d = dst[e];
    const float norm = dinv[s] * ew[e % N_EDGES] * dinv[d];
    if (norm == 0.0f) return;
    const float4 v = *reinterpret_cast<const float4*>(h1 + (size_t)s * HID + j4);
    float* hd = hacc + (size_t)d * HID + j4;
    atomicAdd(hd + 0, v.x * norm);
    atomicAdd(hd + 1, v.y * norm);
    atomicAdd(hd + 2, v.z * norm);
    atomicAdd(hd + 3, v.w * norm);
}

// ---- 6) bias + relu: h2 = relu(hacc + b1) (reuses h1 buffer) ---------------
__global__ void relu_bias_kernel(const float* __restrict__ hacc,
                                 const float* __restrict__ b1,
                                 float* __restrict__ h2) {
    int i = blockIdx.x * blockDim.x + threadIdx.x;
    if (i < HN) {
        float v = hacc[i] + b1[i & (HID - 1)];
        h2[i] = fmaxf(v, 0.0f);
    }
}

// ---- 7) readout GEMM: g[64,32] = h2.reshape(64,128000) @ Wn ----------------
// Each wave owns a distinct K-chunk and computes the FULL 64x32 output block
// (4 M-subtiles x 2 N-subtiles = 8 accumulators): every A fragment feeds 2
// WMMAs, every B fragment feeds 4; no operand element is loaded twice.
constexpr int G2_BLOCKS = 25;
constexpr int G2_CHUNKS = G2_BLOCKS * 8;       // 200 wave-chunks
constexpr int G2_KCHUNK = KREAD / G2_CHUNKS;   // 640 (multiple of 4)
__global__ void __launch_bounds__(256)
gemm2_wmma(const float* __restrict__ h2,   // [64, 128000]
           const float* __restrict__ Wn,   // [128000, 32]
           float* __restrict__ gacc) {     // [64, 32]
    const int lane  = threadIdx.x;
    const int wave  = threadIdx.y;
    const int chunk = blockIdx.x * 8 + wave;
    const int kbase = chunk * G2_KCHUNK;
    const int l16   = lane & 15;
    const int half  = lane >> 4;

    v8f c[4][2] = {};
    for (int k0 = kbase; k0 < kbase + G2_KCHUNK; k0 += 4) {
        const int kb = k0 + half * 2;
        __builtin_prefetch(&Wn[(size_t)(kb + 64) * 32], 0, 1);
        v2f b0, b1;
        b0.x = Wn[(size_t)kb       * 32 + l16];        // N tile 0
        b0.y = Wn[(size_t)(kb + 1) * 32 + l16];
        b1.x = Wn[(size_t)kb       * 32 + 16 + l16];   // N tile 1
        b1.y = Wn[(size_t)(kb + 1) * 32 + 16 + l16];
        #pragma unroll
        for (int mt = 0; mt < 4; ++mt) {
            const float2 av = *reinterpret_cast<const float2*>(
                &h2[(size_t)(mt * 16 + l16) * KREAD + kb]);
            v2f a;
            a.x = av.x;
            a.y = av.y;
            c[mt][0] = __builtin_amdgcn_wmma_f32_16x16x4_f32(
                           false, a, false, b0, (short)0, c[mt][0], false, false);
            c[mt][1] = __builtin_amdgcn_wmma_f32_16x16x4_f32(
                           false, a, false, b1, (short)0, c[mt][1], false, false);
        }
    }

    #pragma unroll
    for (int mt = 0; mt < 4; ++mt) {
        #pragma unroll
        for (int nt = 0; nt < 2; ++nt) {
            #pragma unroll
            for (int v = 0; v < 8; ++v) {
                const int Mr = mt * 16 + v + half * 8;
                const int Nc = nt * 16 + l16;
                atomicAdd(&gacc[Mr * 32 + Nc], c[mt][nt][v]);
            }
        }
    }
}

// ---- 8) classifier: out[64,2] = (g + bn) @ Wc + bc -------------------------
__global__ void final_kernel(const float* __restrict__ gacc,
                             const float* __restrict__ bn,
                             const float* __restrict__ Wc,
                             const float* __restrict__ bc,
                             float* __restrict__ out) {
    int t = blockIdx.x * blockDim.x + threadIdx.x;
    if (t < BATCH * N_CLASSES) {
        const int b  = t >> 1;
        const int cc = t & 1;
        float acc = bc[cc];
        #pragma unroll
        for (int j = 0; j < 32; ++j)
            acc += (gacc[b * 32 + j] + bn[j]) * Wc[j * N_CLASSES + cc];
        out[t] = acc;
    }
}

// ---- host launcher ---------------------------------------------------------
extern "C" void kernel_launch(void* const* d_in, const int* in_sizes, int n_in,
                              void* d_out, int out_size, void* d_ws, size_t ws_size,
                              hipStream_t stream) {
    const float* x     = (const float*)d_in[0];
    const int*   ei    = (const int*)  d_in[1];
    /* d_in[2] = batch (unused: graphs are contiguous) */
    const float* ew_in = (const float*)d_in[3];
    const float* ni_in = (const float*)d_in[4];
    const float* W1    = (const float*)d_in[5];
    const float* b1    = (const float*)d_in[6];
    const float* Wn    = (const float*)d_in[7];
    const float* bn    = (const float*)d_in[8];
    const float* Wc    = (const float*)d_in[9];
    const float* bc    = (const float*)d_in[10];

    const int* src = ei;
    const int* dst = ei + ET;

    float* out    = (float*)d_out;
    float* out_ew = out + BATCH * N_CLASSES;
    float* out_ni = out_ew + N_EDGES;

    float* ws   = (float*)d_ws;
    float* deg  = ws + WS_DEG;
    float* dinv = ws + WS_DINV;
    float* h1   = ws + WS_H1;
    float* hacc = ws + WS_HACC;
    float* gacc = ws + WS_GACC;

    prep_kernel<<<(NT + 255) / 256, 256, 0, stream>>>(ew_in, ni_in, out_ew, out_ni, deg, gacc);
    deg_kernel<<<(ET + 255) / 256, 256, 0, stream>>>(src, dst, out_ew, deg);
    dinv_kernel<<<(NT + 255) / 256, 256, 0, stream>>>(deg, dinv);
    gemm1_wmma<<<NT / G1_MT, dim3(32, 8), 0, stream>>>(x, ni_in, W1, dinv, h1, hacc);
    {
        long threads = (long)ET * 32;
        scatter_kernel<<<(unsigned)((threads + 255) / 256), 256, 0, stream>>>(
            src, dst, out_ew, dinv, h1, hacc);
    }
    relu_bias_kernel<<<(HN + 255) / 256, 256, 0, stream>>>(hacc, b1, h1);
    gemm2_wmma<<<G2_BLOCKS, dim3(32, 8), 0, stream>>>(h1, Wn, gacc);
    final_kernel<<<1, 128, 0, stream>>>(gacc, bn, Wc, bc, out);
}